// PCEN_89567247990959
// MI455X (gfx1250) — compile-verified
//
#include <hip/hip_runtime.h>

// PCEN: out = (x / (FLOOR + ema(x))^a + delta)^(1/r) - delta^(1/r)
// Strategy: chunked parallel linear-recurrence scan; in-chunk reconstruction
// done as 16x16 lower-triangular matmuls on the WMMA pipe (V_WMMA_F32_16X16X4_F32).

#define SMOOTH_F 0.04f
#define P_F      0.96f
#define FLOOR_F  1e-6f

constexpr int kB   = 32;
constexpr int kT   = 4096;
constexpr int kC   = 128;
constexpr int kL   = 128;          // timesteps per chunk
constexpr int kNCH = kT / kL;      // 32 chunks per (b,c) sequence

typedef __attribute__((ext_vector_type(2))) float v2f;
typedef __attribute__((ext_vector_type(8))) float v8f;

__device__ __forceinline__ float fast_pow(float u, float e) {
    // u > 0 in all uses here (FLOOR+m > 0, delta = 2 > 0)
    return __expf(e * __logf(u));
}

__device__ __forceinline__ float ipowp(int k) {
    // P_F^k for small non-negative integer k, exact-ish via square&multiply
    float r = 1.0f, b = P_F;
    while (k) { if (k & 1) r *= b; b *= b; k >>= 1; }
    return r;
}

// ---------------------------------------------------------------------------
// Pass 1: per-(b, chunk, c) EMA endpoint with zero carry-in.
// grid = kB*kNCH blocks x kC threads; each wave's loads are 128B-coalesced.
// ---------------------------------------------------------------------------
__global__ void __launch_bounds__(kC)
pcen_chunk_end(const float* __restrict__ x, float* __restrict__ E) {
    const int b = blockIdx.x / kNCH;
    const int q = blockIdx.x % kNCH;
    const int c = threadIdx.x;
    const size_t base = ((size_t)b * kT + (size_t)q * kL) * kC + c;

    float e = 0.0f;
    #pragma unroll 8
    for (int t = 0; t < kL; ++t) {
        if (t + 16 < kL)
            __builtin_prefetch(&x[base + (size_t)(t + 16) * kC], 0, 1);
        const float xv = x[base + (size_t)t * kC];
        e = fmaf(P_F, e, SMOOTH_F * xv);       // e = s*x + p*e
    }
    E[((size_t)b * kNCH + q) * kC + c] = e;
}

// ---------------------------------------------------------------------------
// Pass 2: tiny cross-chunk scan per (b,c). Overwrites E[q] with the carry-in
// (m at the end of chunk q-1; m_{-1} = x[b,0,c] since the reference seeds
// m0 = x[:,0,:], making m_0 == x_0).
// ---------------------------------------------------------------------------
__global__ void __launch_bounds__(kC)
pcen_scan_carry(const float* __restrict__ x, float* __restrict__ E) {
    const int b = blockIdx.x;
    const int c = threadIdx.x;
    const float dL = ipowp(kL);                // p^L
    float m = x[(size_t)b * kT * kC + c];      // m_{-1}
    #pragma unroll 1
    for (int q = 0; q < kNCH; ++q) {
        const size_t idx = ((size_t)b * kNCH + q) * kC + c;
        const float e = E[idx];
        E[idx] = m;                            // carry-in for chunk q
        m = fmaf(dL, m, e);                    // m_end(q) = E_q + p^L * m_end(q-1)
    }
}

// ---------------------------------------------------------------------------
// Pass 3: reconstruct m inside each chunk with WMMA and apply the PCEN
// nonlinearity. Block = 256 threads = 8 waves; wave w owns channels
// [16w, 16w+16). For each 16-timestep group:
//   m(16t x 16c) = L(16x16, lower-tri s*p^(i-j)) * X(16t x 16c)
//                + p^(t+1) (x) carry(16c)
// computed as 4 chained V_WMMA_F32_16X16X4_F32 (K split 4x4) with the carry
// rank-1 term folded into the accumulator init. Carry for the next group is
// row 15 of D (VGPR 7, lanes 16..31), broadcast with a wave shuffle.
// ---------------------------------------------------------------------------
__global__ void __launch_bounds__(256)
pcen_apply(const float* __restrict__ x,
           const float* __restrict__ alpha,
           const float* __restrict__ delta,
           const float* __restrict__ root,
           const float* __restrict__ carryBuf,
           float* __restrict__ out) {
    const int b    = blockIdx.x / kNCH;
    const int q    = blockIdx.x % kNCH;
    const int lane = threadIdx.x & 31;
    const int w    = threadIdx.x >> 5;
    const int i    = lane & 15;                // N (channel-in-tile) / M (A-row)
    const int hi   = lane >> 4;                // half-wave select
    const int c    = (w << 4) + i;             // absolute channel

    // Per-channel parameters
    const float a_c  = fminf(alpha[c], 1.0f);
    const float d_c  = delta[c];
    const float ir_c = 1.0f / fmaxf(root[c], 1.0f);
    const float dpow = fast_pow(d_c, ir_c);

    float carry = carryBuf[((size_t)b * kNCH + q) * kC + c];

    // A-matrix (16x4 f32 layout: VGPR0 -> K = 2*hi, VGPR1 -> K = 2*hi+1),
    // four K-slices of the lower-triangular EMA weight matrix.
    v2f A[4];
    #pragma unroll
    for (int kk = 0; kk < 4; ++kk) {
        const int j0 = 4 * kk + 2 * hi;
        const int j1 = j0 + 1;
        A[kk].x = (j0 <= i) ? SMOOTH_F * ipowp(i - j0) : 0.0f;
        A[kk].y = (j1 <= i) ? SMOOTH_F * ipowp(i - j1) : 0.0f;
    }
    // Carry decay per accumulator row: D row m = r + 8*hi gets p^(m+1)*carry.
    float pm1[8];
    #pragma unroll
    for (int r = 0; r < 8; ++r) pm1[r] = ipowp(r + 8 * hi + 1);

    const size_t rowBase = (size_t)b * kT * kC;

    #pragma unroll 1
    for (int g = 0; g < kL / 16; ++g) {
        const int t0 = q * kL + g * 16;
        const size_t gbase = rowBase + (size_t)t0 * kC + c;

        // B-matrix tile (4x16 f32: VGPR0 -> K = 2*hi, VGPR1 -> K = 2*hi+1)
        v2f Bt[4];
        #pragma unroll
        for (int kk = 0; kk < 4; ++kk) {
            const int k0 = 4 * kk + 2 * hi;
            Bt[kk].x = x[gbase + (size_t)k0 * kC];
            Bt[kk].y = x[gbase + (size_t)(k0 + 1) * kC];
        }

        // Accumulator init = rank-1 carry term
        v8f acc;
        #pragma unroll
        for (int r = 0; r < 8; ++r) acc[r] = pm1[r] * carry;

        // m = L * X + carry-term, K accumulated 4 at a time on the matrix pipe
        acc = __builtin_amdgcn_wmma_f32_16x16x4_f32(false, A[0], false, Bt[0],
                                                    (short)0, acc, false, false);
        acc = __builtin_amdgcn_wmma_f32_16x16x4_f32(false, A[1], false, Bt[1],
                                                    (short)0, acc, false, false);
        acc = __builtin_amdgcn_wmma_f32_16x16x4_f32(false, A[2], false, Bt[2],
                                                    (short)0, acc, false, false);
        acc = __builtin_amdgcn_wmma_f32_16x16x4_f32(false, A[3], false, Bt[3],
                                                    (short)0, acc, false, false);

        // Next-group carry = m at t0+15 for this lane's channel:
        // lives in acc[7] of lane 16+i (M=15, N=i).
        carry = __shfl(acc[7], 16 + i, 32);

        // Apply PCEN nonlinearity; x re-read hits WGP$/L2 (x fits in 192MB L2).
        #pragma unroll
        for (int r = 0; r < 8; ++r) {
            const int t = t0 + r + 8 * hi;            // D row -> timestep
            const size_t idx = rowBase + (size_t)t * kC + c;
            const float xv = x[idx];
            const float m  = acc[r];
            const float y  = fast_pow(xv / fast_pow(FLOOR_F + m, a_c) + d_c, ir_c) - dpow;
            out[idx] = y;
        }
    }
}

// ---------------------------------------------------------------------------
extern "C" void kernel_launch(void* const* d_in, const int* in_sizes, int n_in,
                              void* d_out, int out_size, void* d_ws, size_t ws_size,
                              hipStream_t stream) {
    (void)in_sizes; (void)n_in; (void)out_size; (void)ws_size;
    const float* x     = (const float*)d_in[0];
    const float* alpha = (const float*)d_in[1];
    const float* delta = (const float*)d_in[2];
    const float* root  = (const float*)d_in[3];
    float* out = (float*)d_out;
    float* E   = (float*)d_ws;   // kB*kNCH*kC floats = 512 KB scratch

    pcen_chunk_end <<<kB * kNCH, kC,  0, stream>>>(x, E);
    pcen_scan_carry<<<kB,        kC,  0, stream>>>(x, E);
    pcen_apply     <<<kB * kNCH, 256, 0, stream>>>(x, alpha, delta, root, E, out);
}